// SelectiveStructuredSSM_44487271252582
// MI455X (gfx1250) — compile-verified
//
#include <hip/hip_runtime.h>
#include <math.h>

// Problem constants (from reference): B_=2, D=768, L=1024, N=16
#define DD 768
#define LL 1024
#define NN 16
#define BB 2
#define NC 16          // chunks per sequence
#define LC (LL / NC)   // 64 steps per chunk

typedef float v2f __attribute__((ext_vector_type(2)));
typedef float v8f __attribute__((ext_vector_type(8)));

// ---------------------------------------------------------------------------
// Kernel 1: fused projection GEMMs on the WMMA pipe (full fp32 precision).
//   Bp = xT @ W_B^T + b_B   (M=B*L=2048, K=768, N=16)
//   Cp = xT @ W_C^T + b_C
//   dlin = xT @ W_d^T + b_d (column 0 of a third accumulator)
// One wave per 16-row tile; 128 tiles. v_wmma_f32_16x16x4_f32, K-loop of 192.
// ---------------------------------------------------------------------------
__global__ __launch_bounds__(32) void ssm_proj_wmma(
    const float* __restrict__ x,    // (B, D, L)
    const float* __restrict__ W_B,  // (N, D)
    const float* __restrict__ b_B,  // (N,)
    const float* __restrict__ W_C,  // (N, D)
    const float* __restrict__ b_C,  // (N,)
    const float* __restrict__ W_d,  // (1, D)
    const float* __restrict__ b_d,  // (1,)
    float* __restrict__ Bp,         // (B, L, N)
    float* __restrict__ Cp,         // (B, L, N)
    float* __restrict__ dlin)       // (B, L)
{
    const int lane = threadIdx.x;      // 0..31
    const int half = lane >> 4;        // which K pair this lane carries
    const int mn   = lane & 15;        // M for A-tile, N for B-tile
    const int row0 = blockIdx.x << 4;  // first global row (over B*L)
    const int b    = row0 >> 10;
    const int l0   = row0 & (LL - 1);
    const int lA   = l0 + mn;          // this lane's A-matrix row (= l index)

    const float biasB = b_B[mn];
    const float biasC = b_C[mn];
    const float biasD = (mn == 0) ? b_d[0] : 0.0f;
    v8f accB, accC, accD;
#pragma unroll
    for (int r = 0; r < 8; ++r) { accB[r] = biasB; accC[r] = biasC; accD[r] = biasD; }

    const float* __restrict__ xb = x + (size_t)b * DD * LL;
    const float n0sel = (mn == 0) ? 1.0f : 0.0f;   // value-select, no EXEC branching

    for (int k0 = 0; k0 < DD; k0 += 4) {
        const int k = k0 + 2 * half;
        v2f a;
        a[0] = xb[(size_t)k * LL + lA];
        a[1] = xb[(size_t)(k + 1) * LL + lA];
        v2f wb, wc, wd;
        wb[0] = W_B[mn * DD + k];
        wb[1] = W_B[mn * DD + k + 1];
        wc[0] = W_C[mn * DD + k];
        wc[1] = W_C[mn * DD + k + 1];
        wd[0] = W_d[k] * n0sel;        // broadcast load, masked by multiply
        wd[1] = W_d[k + 1] * n0sel;

        accB = __builtin_amdgcn_wmma_f32_16x16x4_f32(false, a, false, wb, (short)0, accB, false, false);
        accC = __builtin_amdgcn_wmma_f32_16x16x4_f32(false, a, false, wc, (short)0, accC, false, false);
        accD = __builtin_amdgcn_wmma_f32_16x16x4_f32(false, a, false, wd, (short)0, accD, false, false);
    }

#pragma unroll
    for (int r = 0; r < 8; ++r) {
        const int l  = l0 + r + 8 * half;
        const int gr = b * LL + l;
        Bp[gr * NN + mn] = accB[r];
        Cp[gr * NN + mn] = accC[r];
        if (mn == 0) dlin[gr] = accD[r];
    }
}

// ---------------------------------------------------------------------------
// Kernel 2: delta = softplus(dlin[b,l] + p_Delta[d]), computed once per
// (b,d,l) so the 16 state lanes never redo the transcendental-heavy softplus.
// ---------------------------------------------------------------------------
__global__ __launch_bounds__(256) void ssm_delta(
    const float* __restrict__ dlin,     // (B, L)
    const float* __restrict__ p_Delta,  // (D,)
    float* __restrict__ delta)          // (B, D, L)
{
    const int i  = blockIdx.x * 256 + threadIdx.x;  // over B*D*L
    const int l  = i & (LL - 1);
    const int bd = i >> 10;
    const int b  = bd / DD;
    const int d  = bd - b * DD;
    const float dl = dlin[b * LL + l] + p_Delta[d];
    delta[i] = fmaxf(dl, 0.0f) + log1pf(__expf(-fabsf(dl)));
}

// ---------------------------------------------------------------------------
// Kernel 3 (pass A): chunk-local scans. Each lane owns 4 states (n = 4q..4q+3)
// of one (b,d,chunk); Bp rows load as coalesced float4 (b128).
// Emits P = prod(a) = exp(A * sum(delta))  (one exp per chunk, not per step)
// and  S = local scan end with h_init = 0.
// ---------------------------------------------------------------------------
__global__ __launch_bounds__(256) void ssm_scan_local(
    const float* __restrict__ x,      // (B, D, L)
    const float* __restrict__ A,      // (D, N)
    const float* __restrict__ delta,  // (B, D, L)
    const float* __restrict__ Bp,     // (B, L, N)
    float* __restrict__ Pc,           // (B*D*NC, N)
    float* __restrict__ Sc)           // (B*D*NC, N)
{
    const int gtid  = blockIdx.x * 256 + threadIdx.x;
    const int q     = gtid & 3;
    const int group = gtid >> 2;           // 0 .. B*D*NC-1
    const int c     = group & (NC - 1);
    const int pair  = group >> 4;          // (b*DD + d)
    const int b     = pair / DD;
    const int d     = pair - b * DD;

    const float4 An  = ((const float4*)(A + d * NN))[q];
    const float4 rAn = make_float4(1.0f / An.x, 1.0f / An.y, 1.0f / An.z, 1.0f / An.w);

    const float*  __restrict__ dl  = delta + (size_t)pair * LL + c * LC;
    const float*  __restrict__ xr  = x     + (size_t)pair * LL + c * LC;
    const float4* __restrict__ bpr = (const float4*)(Bp + ((size_t)b * LL + c * LC) * NN) + q;

    float4 h = make_float4(0.f, 0.f, 0.f, 0.f);
    float sd = 0.0f;
#pragma unroll 2
    for (int s = 0; s < LC; ++s) {
        const float dt = dl[s];
        const float xv = xr[s];
        const float4 bp = bpr[4 * s];
        sd += dt;
        const float a0 = __expf(dt * An.x);
        const float a1 = __expf(dt * An.y);
        const float a2 = __expf(dt * An.z);
        const float a3 = __expf(dt * An.w);
        h.x = fmaf(a0, h.x, (a0 - An.x) * rAn.x * bp.x * xv);
        h.y = fmaf(a1, h.y, (a1 - An.y) * rAn.y * bp.y * xv);
        h.z = fmaf(a2, h.z, (a2 - An.z) * rAn.z * bp.z * xv);
        h.w = fmaf(a3, h.w, (a3 - An.w) * rAn.w * bp.w * xv);
    }
    float4 P;
    P.x = __expf(An.x * sd);
    P.y = __expf(An.y * sd);
    P.z = __expf(An.z * sd);
    P.w = __expf(An.w * sd);
    ((float4*)Pc)[group * 4 + q] = P;
    ((float4*)Sc)[group * 4 + q] = h;
}

// ---------------------------------------------------------------------------
// Kernel 4 (pass B): serial combine over the 16 chunks per (b,d,n) channel.
// h_init[c] = combination of chunks 0..c-1.  24576 threads, 16 steps each.
// ---------------------------------------------------------------------------
__global__ __launch_bounds__(256) void ssm_chunk_combine(
    const float* __restrict__ Pc,   // (B*D*NC, N)
    const float* __restrict__ Sc,   // (B*D*NC, N)
    float* __restrict__ Hinit)      // (B*D*NC, N)
{
    const int t    = blockIdx.x * 256 + threadIdx.x;  // 0 .. B*D*N-1
    const int n    = t & 15;
    const int pair = t >> 4;
    float h = 0.0f;
#pragma unroll
    for (int c = 0; c < NC; ++c) {
        const int idx = (pair * NC + c) * NN + n;
        Hinit[idx] = h;
        h = fmaf(Pc[idx], h, Sc[idx]);
    }
}

// ---------------------------------------------------------------------------
// Kernel 5 (pass C): chunk-local scans again, now seeded with Hinit, fused
// with the y = sum_n Cp*h output (4 fma + two width-4 shuffle-adds per step).
// ---------------------------------------------------------------------------
__global__ __launch_bounds__(256) void ssm_scan_final(
    const float* __restrict__ x,      // (B, D, L)
    const float* __restrict__ A,      // (D, N)
    const float* __restrict__ delta,  // (B, D, L)
    const float* __restrict__ Bp,     // (B, L, N)
    const float* __restrict__ Cp,     // (B, L, N)
    const float* __restrict__ Hinit,  // (B*D*NC, N)
    float* __restrict__ y)            // (B, D, L)
{
    const int gtid  = blockIdx.x * 256 + threadIdx.x;
    const int q     = gtid & 3;
    const int group = gtid >> 2;
    const int c     = group & (NC - 1);
    const int pair  = group >> 4;
    const int b     = pair / DD;
    const int d     = pair - b * DD;

    const float4 An  = ((const float4*)(A + d * NN))[q];
    const float4 rAn = make_float4(1.0f / An.x, 1.0f / An.y, 1.0f / An.z, 1.0f / An.w);

    const float*  __restrict__ dl  = delta + (size_t)pair * LL + c * LC;
    const float*  __restrict__ xr  = x     + (size_t)pair * LL + c * LC;
    const float4* __restrict__ bpr = (const float4*)(Bp + ((size_t)b * LL + c * LC) * NN) + q;
    const float4* __restrict__ cpr = (const float4*)(Cp + ((size_t)b * LL + c * LC) * NN) + q;
    float*        __restrict__ yr  = y     + (size_t)pair * LL + c * LC;

    float4 h = ((const float4*)Hinit)[group * 4 + q];

#pragma unroll 2
    for (int s = 0; s < LC; ++s) {
        const float dt = dl[s];
        const float xv = xr[s];
        const float4 bp = bpr[4 * s];
        const float4 cp = cpr[4 * s];
        const float a0 = __expf(dt * An.x);
        const float a1 = __expf(dt * An.y);
        const float a2 = __expf(dt * An.z);
        const float a3 = __expf(dt * An.w);
        h.x = fmaf(a0, h.x, (a0 - An.x) * rAn.x * bp.x * xv);
        h.y = fmaf(a1, h.y, (a1 - An.y) * rAn.y * bp.y * xv);
        h.z = fmaf(a2, h.z, (a2 - An.z) * rAn.z * bp.z * xv);
        h.w = fmaf(a3, h.w, (a3 - An.w) * rAn.w * bp.w * xv);

        float p = cp.x * h.x;
        p = fmaf(cp.y, h.y, p);
        p = fmaf(cp.z, h.z, p);
        p = fmaf(cp.w, h.w, p);
        p += __shfl_xor(p, 1, 4);
        p += __shfl_xor(p, 2, 4);
        if (q == 0) yr[s] = p;
    }
}

// ---------------------------------------------------------------------------
extern "C" void kernel_launch(void* const* d_in, const int* in_sizes, int n_in,
                              void* d_out, int out_size, void* d_ws, size_t ws_size,
                              hipStream_t stream)
{
    const float* x       = (const float*)d_in[0]; // (B, D, L)
    const float* A       = (const float*)d_in[1]; // (D, N)
    const float* W_B     = (const float*)d_in[2];
    const float* b_B     = (const float*)d_in[3];
    const float* W_C     = (const float*)d_in[4];
    const float* b_C     = (const float*)d_in[5];
    const float* W_d     = (const float*)d_in[6];
    const float* b_d     = (const float*)d_in[7];
    const float* p_Delta = (const float*)d_in[8];
    float* y = (float*)d_out;                     // (B, D, L)

    // Workspace carve-up (floats, all offsets 16B-aligned):
    //   Bp 32768 | Cp 32768 | dlin 2048 | delta 1572864 | Pc/Sc/Hinit 393216 ea
    float* ws    = (float*)d_ws;
    float* Bp    = ws;
    float* Cp    = Bp + (size_t)BB * LL * NN;       //  32768
    float* dlin  = Cp + (size_t)BB * LL * NN;       //  65536
    float* delta = dlin + (size_t)BB * LL;          //  67584
    float* Pc    = delta + (size_t)BB * DD * LL;    //  1640448
    float* Sc    = Pc + (size_t)BB * DD * NC * NN;  //  2033664
    float* Hinit = Sc + (size_t)BB * DD * NC * NN;  //  2426880  (~11.3 MB total)

    // 1) WMMA projections: 2048 rows / 16 = 128 single-wave tiles
    ssm_proj_wmma<<<dim3(128), dim3(32), 0, stream>>>(
        x, W_B, b_B, W_C, b_C, W_d, b_d, Bp, Cp, dlin);

    // 2) delta precompute: B*D*L / 256 = 6144 blocks
    ssm_delta<<<dim3((BB * DD * LL) / 256), dim3(256), 0, stream>>>(
        dlin, p_Delta, delta);

    // 3) pass A: B*D*NC groups * 4 lanes = 98304 threads
    ssm_scan_local<<<dim3((BB * DD * NC * 4) / 256), dim3(256), 0, stream>>>(
        x, A, delta, Bp, Pc, Sc);

    // 4) pass B: B*D*N = 24576 threads
    ssm_chunk_combine<<<dim3((BB * DD * NN) / 256), dim3(256), 0, stream>>>(
        Pc, Sc, Hinit);

    // 5) pass C: seeded local scans + fused output projection
    ssm_scan_final<<<dim3((BB * DD * NC * 4) / 256), dim3(256), 0, stream>>>(
        x, A, delta, Bp, Cp, Hinit, y);
}